// MultiTaskAffineLayer_9156870275141
// MI455X (gfx1250) — compile-verified
//
#include <hip/hip_runtime.h>

#define IN_CHANS   256
#define OUT_CHANS  256
#define N_TASKS    64
#define W_SIZE     (OUT_CHANS * IN_CHANS)     // 65536
#define FULL_EMBED (W_SIZE + OUT_CHANS)       // 65792
#define MAX_BATCH  4096
#define TILE_M     16

typedef __attribute__((ext_vector_type(2))) float v2f;
typedef __attribute__((ext_vector_type(8))) float v8f;

// ---------------------------------------------------------------------------
// Kernel 0: zero per-task counters (must run every call: graph replays don't
// re-poison d_ws).
// ---------------------------------------------------------------------------
__global__ void mt_bucket_zero(int* __restrict__ counts) {
    if (threadIdx.x < N_TASKS) counts[threadIdx.x] = 0;
}

// ---------------------------------------------------------------------------
// Kernel 1: bucket batch rows by task id.  lists[t][j] = j-th batch row of
// task t.  Order within a bucket is non-deterministic but the final output
// values are order-independent.
// ---------------------------------------------------------------------------
__global__ void mt_bucket_fill(const int* __restrict__ task_ids,
                               int* __restrict__ counts,
                               int* __restrict__ lists, int B) {
    int b = blockIdx.x * blockDim.x + threadIdx.x;
    if (b < B) {
        int t = task_ids[b];
        int pos = atomicAdd(&counts[t], 1);
        lists[t * MAX_BATCH + pos] = b;
    }
}

// ---------------------------------------------------------------------------
// Kernel 2: per (task, 16-row tile) workgroup.  8 waves x 32 out-cols = 256
// out-cols.  D(16x16) = A(16x4 = gathered X rows) x B(4x16 = W^T) accumulated
// over K=256 in steps of 4 with V_WMMA_F32_16X16X4_F32 (exact fp32).
//
// VGPR layouts per CDNA5 ISA 7.12.2 (32-bit A 16x4 / B 4x16):
//   lane l, hi = l>>4, n = l&15
//   A: reg{0,1} = X[row n][kbase + 2*hi + {0,1}]
//   B: reg{0,1} = W[o0 + n][kbase + 2*hi + {0,1}]
//   D: vgpr r -> M = r + 8*hi (batch row), N = n (out col)
//
// The K loop is software-pipelined one iteration deep so the global_load_b64s
// for step kk+1 are in flight while the two WMMAs of step kk execute.
// ---------------------------------------------------------------------------
__global__ __launch_bounds__(256)
void mt_affine_wmma(const float* __restrict__ x,
                    const float* __restrict__ emb,
                    const int*   __restrict__ counts,
                    const int*   __restrict__ lists,
                    float*       __restrict__ out) {
    const int t    = blockIdx.x;
    const int row0 = blockIdx.y * TILE_M;
    const int cnt  = counts[t];
    if (row0 >= cnt) return;                       // block-uniform exit
    const int rows_valid = min(TILE_M, cnt - row0);

    const int lane = threadIdx.x & 31;
    const int wave = threadIdx.x >> 5;
    const int hi   = lane >> 4;                    // half-wave select
    const int n    = lane & 15;

    // Gathered batch-row index for tile row n (clamped for the tail tile;
    // garbage rows only affect D rows that are never stored).
    const int* list = lists + t * MAX_BATCH + row0;
    const int  bi   = list[(n < rows_valid) ? n : 0];

    const float* wbase = emb + (size_t)t * FULL_EMBED;
    const int    o0    = wave * 32;

    const v2f* __restrict__ xrow  = (const v2f*)(x     + (size_t)bi * IN_CHANS);
    const v2f* __restrict__ wrow0 = (const v2f*)(wbase + (size_t)(o0 + n)      * IN_CHANS);
    const v2f* __restrict__ wrow1 = (const v2f*)(wbase + (size_t)(o0 + 16 + n) * IN_CHANS);

    v8f c0 = {};
    v8f c1 = {};

    // Software pipeline: fragments for step kk are loaded during step kk-1.
    v2f a  = xrow [hi];
    v2f b0 = wrow0[hi];
    v2f b1 = wrow1[hi];

    // kk counts in float2 units; K advances by 4 floats per iteration.
    #pragma unroll 8
    for (int kk = 2; kk < IN_CHANS / 2; kk += 2) {
        v2f an  = xrow [kk + hi];
        v2f b0n = wrow0[kk + hi];
        v2f b1n = wrow1[kk + hi];
        c0 = __builtin_amdgcn_wmma_f32_16x16x4_f32(false, a, false, b0,
                                                   (short)0, c0, false, false);
        c1 = __builtin_amdgcn_wmma_f32_16x16x4_f32(false, a, false, b1,
                                                   (short)0, c1, false, false);
        a = an; b0 = b0n; b1 = b1n;
    }
    c0 = __builtin_amdgcn_wmma_f32_16x16x4_f32(false, a, false, b0,
                                               (short)0, c0, false, false);
    c1 = __builtin_amdgcn_wmma_f32_16x16x4_f32(false, a, false, b1,
                                               (short)0, c1, false, false);

    const float bias0 = wbase[W_SIZE + o0 + n];
    const float bias1 = wbase[W_SIZE + o0 + 16 + n];

    // Store: vgpr r -> batch row bi[r + 8*hi]; lanes of a half-wave write 16
    // consecutive floats (64B segment).
    #pragma unroll
    for (int r = 0; r < 8; ++r) {
        int   m   = r + hi * 8;
        int   row = __shfl(bi, m, 32);
        if (m < rows_valid) {
            float* orow = out + (size_t)row * OUT_CHANS;
            orow[o0 + n]      = c0[r] + bias0;
            orow[o0 + 16 + n] = c1[r] + bias1;
        }
    }
}

// ---------------------------------------------------------------------------
// Fallback (no scratch): direct per-row matvec, correct but ~40x more HBM
// traffic.  Only used if d_ws is too small or B > MAX_BATCH.
// ---------------------------------------------------------------------------
__global__ void mt_affine_naive(const float* __restrict__ x,
                                const int*   __restrict__ task_ids,
                                const float* __restrict__ emb,
                                float* __restrict__ out, int B) {
    int b = blockIdx.x;
    int o = threadIdx.x;
    if (b >= B || o >= OUT_CHANS) return;
    int t = task_ids[b];
    const float* w  = emb + (size_t)t * FULL_EMBED + (size_t)o * IN_CHANS;
    const float* xi = x   + (size_t)b * IN_CHANS;
    float acc = emb[(size_t)t * FULL_EMBED + W_SIZE + o];
    for (int i = 0; i < IN_CHANS; ++i) acc += w[i] * xi[i];
    out[(size_t)b * OUT_CHANS + o] = acc;
}

extern "C" void kernel_launch(void* const* d_in, const int* in_sizes, int n_in,
                              void* d_out, int out_size, void* d_ws, size_t ws_size,
                              hipStream_t stream) {
    const float* x        = (const float*)d_in[0];
    const int*   task_ids = (const int*)  d_in[1];
    const float* emb      = (const float*)d_in[2];
    float*       out      = (float*)d_out;
    const int B = in_sizes[1];

    const size_t need = (size_t)(N_TASKS + N_TASKS * MAX_BATCH) * sizeof(int);
    if (B <= MAX_BATCH && ws_size >= need) {
        int* counts = (int*)d_ws;
        int* lists  = counts + N_TASKS;
        mt_bucket_zero<<<1, N_TASKS, 0, stream>>>(counts);
        mt_bucket_fill<<<(B + 255) / 256, 256, 0, stream>>>(task_ids, counts, lists, B);
        dim3 grid(N_TASKS, (MAX_BATCH + TILE_M - 1) / TILE_M);
        mt_affine_wmma<<<grid, 256, 0, stream>>>(x, emb, counts, lists, out);
    } else {
        mt_affine_naive<<<B, OUT_CHANS, 0, stream>>>(x, task_ids, emb, out, B);
    }
}